// InvertibleMLPFlow_10514079941377
// MI455X (gfx1250) — compile-verified
//
#include <hip/hip_runtime.h>
#include <hip/hip_fp16.h>

#define LATENT 64
#define HIDDEN 256
#define LAYERS 4
#define BATCH  8192
#define SLOPE  0.01f

#define NWAVE 4      // waves per block in jac kernel
#define BPW   8      // batch elements per wave
#define W2P   257    // padded pitch for W2 (64x256) in LDS
#define W1P   65     // padded pitch for W1 (256x64) in LDS
#define JP    65     // padded pitch for J (64x64) in LDS

typedef float v2f __attribute__((ext_vector_type(2)));
typedef float v8f __attribute__((ext_vector_type(8)));

// Compiler-only memory barrier: LDS ops within a wave complete in order on
// CDNA5 and lanes are lockstep, so we only need to stop compiler reordering.
__device__ __forceinline__ void lds_cb() { asm volatile("" ::: "memory"); }

// ---------------------------------------------------------------------------
// Forward MLP layer: theta_out = lrelu(theta W1^T + b1) W2^T + b2
// Also computes mask from h2 = theta_out W1^T + b1 (per reference).
// ---------------------------------------------------------------------------
__global__ void __launch_bounds__(256, 1)
fwd_kernel(const float* __restrict__ tin,
           const float* __restrict__ W1,
           const float* __restrict__ b1,
           const float* __restrict__ W2,
           const float* __restrict__ b2,
           float* __restrict__ tout,
           float* __restrict__ maskout)
{
    extern __shared__ float smem[];
    float* W1c = smem;                       // 256*64
    float* W2c = W1c + HIDDEN * LATENT;      // 64*256
    float* b1c = W2c + LATENT * HIDDEN;      // 256
    float* b2c = b1c + HIDDEN;               // 64

    const int tid = threadIdx.x;
    for (int i = tid; i < HIDDEN * LATENT; i += blockDim.x) W1c[i] = W1[i];
    for (int i = tid; i < LATENT * HIDDEN; i += blockDim.x) W2c[i] = W2[i];
    for (int i = tid; i < HIDDEN; i += blockDim.x) b1c[i] = b1[i];
    for (int i = tid; i < LATENT; i += blockDim.x) b2c[i] = b2[i];
    __syncthreads();

    const int b = blockIdx.x * blockDim.x + tid;
    float x[LATENT], t[LATENT];
    #pragma unroll
    for (int k = 0; k < LATENT; ++k) x[k] = tin[b * LATENT + k];
    #pragma unroll
    for (int d = 0; d < LATENT; ++d) t[d] = b2c[d];

    #pragma unroll 1
    for (int j = 0; j < HIDDEN; ++j) {
        float h = b1c[j];
        #pragma unroll
        for (int k = 0; k < LATENT; ++k) h = fmaf(x[k], W1c[j * LATENT + k], h);
        float a = (h > 0.f) ? h : SLOPE * h;
        #pragma unroll
        for (int d = 0; d < LATENT; ++d) t[d] = fmaf(a, W2c[d * HIDDEN + j], t[d]);
    }
    #pragma unroll
    for (int d = 0; d < LATENT; ++d) tout[b * LATENT + d] = t[d];

    #pragma unroll 1
    for (int j = 0; j < HIDDEN; ++j) {
        float h = b1c[j];
        #pragma unroll
        for (int k = 0; k < LATENT; ++k) h = fmaf(t[k], W1c[j * LATENT + k], h);
        maskout[b * HIDDEN + j] = (h > 0.f) ? 1.0f : SLOPE;
    }
}

// ---------------------------------------------------------------------------
// Per-sample Jacobian J = W2 diag(mask) W1 via f32 WMMA (16x16x4), then
// wave-synchronous LU with partial pivoting in LDS -> log|det|.
// One wave per sample; 4 waves/block; BPW samples per wave.
// ---------------------------------------------------------------------------
__global__ void __launch_bounds__(128, 1)
jac_kernel(const float* __restrict__ maskg,
           const float* __restrict__ W1,
           const float* __restrict__ W2,
           float* __restrict__ logabs)
{
    extern __shared__ float smem[];
    float* W2s = smem;                        // 64 x 257
    float* W1s = W2s + LATENT * W2P;          // 256 x 65
    float* msk = W1s + HIDDEN * W1P;          // NWAVE x 256
    float* Jb  = msk + NWAVE * HIDDEN;        // NWAVE x (64 x 65)

    const int tid  = threadIdx.x;
    const int lane = tid & 31;
    const int wave = tid >> 5;

    // Stage weights (padded rows for bank-conflict-free fragment gathers)
    for (int i = tid; i < LATENT * HIDDEN; i += NWAVE * 32) {
        int r = i >> 8, c = i & 255;          // W2: 64 x 256
        W2s[r * W2P + c] = W2[i];
    }
    for (int i = tid; i < HIDDEN * LATENT; i += NWAVE * 32) {
        int r = i >> 6, c = i & 63;           // W1: 256 x 64
        W1s[r * W1P + c] = W1[i];
    }
    __syncthreads();

    const int lm    = lane & 15;
    const int khalf = lane >> 4;              // 0 or 1
    float* mw = msk + wave * HIDDEN;
    float* jw = Jb + wave * LATENT * JP;

    // Per-lane invariant LDS index bases (k added inside the loop)
    int aBase0 = (0 * 16 + lm) * W2P + 2 * khalf;
    int aBase1 = (1 * 16 + lm) * W2P + 2 * khalf;
    int aBase2 = (2 * 16 + lm) * W2P + 2 * khalf;
    int aBase3 = (3 * 16 + lm) * W2P + 2 * khalf;
    int bBase  = 2 * khalf * W1P + lm;        // + k0*W1P + n*16

    #pragma unroll 1
    for (int rep = 0; rep < BPW; ++rep) {
        const int b = (blockIdx.x * NWAVE + wave) * BPW + rep;

        // Load this sample's mask row into per-wave LDS
        #pragma unroll
        for (int t = 0; t < 8; ++t)
            mw[lane + 32 * t] = maskg[b * HIDDEN + lane + 32 * t];
        __syncthreads();

        v8f acc[4][4];
        #pragma unroll
        for (int m = 0; m < 4; ++m)
            #pragma unroll
            for (int n = 0; n < 4; ++n)
                #pragma unroll
                for (int v = 0; v < 8; ++v) acc[m][n][v] = 0.f;

        // J = (W2) x (diag(mask) W1), K = 256, step 4
        #pragma unroll 1
        for (int k0 = 0; k0 < HIDDEN; k0 += 4) {
            const int kk = k0 + 2 * khalf;
            const float m0 = mw[kk], m1 = mw[kk + 1];
            v2f a[4], bb[4];
            a[0].x = W2s[aBase0 + k0];  a[0].y = W2s[aBase0 + k0 + 1];
            a[1].x = W2s[aBase1 + k0];  a[1].y = W2s[aBase1 + k0 + 1];
            a[2].x = W2s[aBase2 + k0];  a[2].y = W2s[aBase2 + k0 + 1];
            a[3].x = W2s[aBase3 + k0];  a[3].y = W2s[aBase3 + k0 + 1];
            #pragma unroll
            for (int n = 0; n < 4; ++n) {     // B: 4x16 tile of diag(mask) W1
                bb[n].x = m0 * W1s[bBase + k0 * W1P + n * 16];
                bb[n].y = m1 * W1s[bBase + (k0 + 1) * W1P + n * 16];
            }
            #pragma unroll
            for (int m = 0; m < 4; ++m)
                #pragma unroll
                for (int n = 0; n < 4; ++n)
                    acc[m][n] = __builtin_amdgcn_wmma_f32_16x16x4_f32(
                        false, a[m], false, bb[n], (short)0, acc[m][n],
                        false, false);
        }

        // Spill J tiles to LDS (C/D layout: VGPR v -> rows v / v+8)
        #pragma unroll
        for (int m = 0; m < 4; ++m)
            #pragma unroll
            for (int n = 0; n < 4; ++n)
                #pragma unroll
                for (int v = 0; v < 8; ++v) {
                    int row = m * 16 + v + 8 * khalf;
                    int col = n * 16 + lm;
                    jw[row * JP + col] = acc[m][n][v];
                }
        __syncthreads();

        // ---- LU with partial pivoting (wave-synchronous, lockstep) ----
        float la = 0.f;
        const int j1 = lane, j2 = lane + 32;   // columns owned by this lane
        #pragma unroll 1
        for (int k = 0; k < LATENT; ++k) {
            // argmax |J[i][k]| for i >= k  (shuffle reduce, smallest-index tie)
            float bv = -1.f; int bi = k;
            int i1 = k + lane;
            if (i1 < LATENT) { float v = fabsf(jw[i1 * JP + k]); if (v > bv) { bv = v; bi = i1; } }
            int i2 = i1 + 32;
            if (i2 < LATENT) { float v = fabsf(jw[i2 * JP + k]); if (v > bv) { bv = v; bi = i2; } }
            #pragma unroll
            for (int off = 16; off > 0; off >>= 1) {
                float ov = __shfl_xor(bv, off, 32);
                int   oi = __shfl_xor(bi, off, 32);
                if (ov > bv || (ov == bv && oi < bi)) { bv = ov; bi = oi; }
            }
            lds_cb();
            // swap rows k <-> bi (each lane owns columns j1, j2)
            {
                float t1 = jw[k * JP + j1], s1 = jw[bi * JP + j1];
                jw[k * JP + j1] = s1; jw[bi * JP + j1] = t1;
                float t2 = jw[k * JP + j2], s2 = jw[bi * JP + j2];
                jw[k * JP + j2] = s2; jw[bi * JP + j2] = t2;
            }
            lds_cb();
            float piv = jw[k * JP + k];        // broadcast read
            la += logf(fabsf(piv));
            float inv = 1.0f / piv;
            // Row-k values for this lane's columns, hoisted once per k
            float rk1 = jw[k * JP + j1];
            float rk2 = jw[k * JP + j2];
            const bool u1 = (j1 > k);
            const bool u2 = (j2 > k);
            #pragma unroll 2
            for (int i = k + 1; i < LATENT; ++i) {
                float f = jw[i * JP + k] * inv;      // broadcast read (col k never written)
                if (u1) jw[i * JP + j1] = fmaf(-f, rk1, jw[i * JP + j1]);
                if (u2) jw[i * JP + j2] = fmaf(-f, rk2, jw[i * JP + j2]);
            }
            lds_cb();
        }
        if (lane == 0) logabs[b] = la;
        __syncthreads();
    }
}

// ---------------------------------------------------------------------------
// logsumexp over 8192 log|det| values (rounded through f16 like reference);
// accumulates into the ldj scalar (layer 0 initializes).
// ---------------------------------------------------------------------------
__global__ void lse_kernel(const float* __restrict__ logabs,
                           float* __restrict__ ldj, int accumulate)
{
    __shared__ float red[256];
    const int tid = threadIdx.x;
    float m = -INFINITY;
    for (int i = tid; i < BATCH; i += 256) {
        float v = __half2float(__float2half(logabs[i]));
        m = fmaxf(m, v);
    }
    red[tid] = m; __syncthreads();
    for (int s = 128; s > 0; s >>= 1) {
        if (tid < s) red[tid] = fmaxf(red[tid], red[tid + s]);
        __syncthreads();
    }
    m = red[0]; __syncthreads();
    float su = 0.f;
    for (int i = tid; i < BATCH; i += 256) {
        float v = __half2float(__float2half(logabs[i]));
        su += __expf(v - m);
    }
    red[tid] = su; __syncthreads();
    for (int s = 128; s > 0; s >>= 1) {
        if (tid < s) red[tid] += red[tid + s];
        __syncthreads();
    }
    if (tid == 0) {
        float r = m + logf(red[0]);
        r = __half2float(__float2half(r));   // reference result dtype chain
        ldj[0] = (accumulate ? ldj[0] : 0.f) + r;
    }
}

// ---------------------------------------------------------------------------
__global__ void out_kernel(const float* __restrict__ theta,
                           const float* __restrict__ ldj,
                           float* __restrict__ out)
{
    int i = blockIdx.x * blockDim.x + threadIdx.x;
    if (i < BATCH * LATENT) out[i] = theta[i];
    if (i == 0) out[BATCH * LATENT] = ldj[0];
}

// ---------------------------------------------------------------------------
extern "C" void kernel_launch(void* const* d_in, const int* in_sizes, int n_in,
                              void* d_out, int out_size, void* d_ws, size_t ws_size,
                              hipStream_t stream)
{
    const float* z  = (const float*)d_in[0];   // 8192x64
    const float* W1 = (const float*)d_in[1];   // 4x256x64
    const float* b1 = (const float*)d_in[2];   // 4x256
    const float* W2 = (const float*)d_in[3];   // 4x64x256
    const float* b2 = (const float*)d_in[4];   // 4x64

    float* ws  = (float*)d_ws;
    float* thA = ws;
    float* thB = thA + BATCH * LATENT;
    float* msk = thB + BATCH * LATENT;
    float* lab = msk + BATCH * HIDDEN;
    float* ldj = lab + BATCH;

    const size_t fwd_smem = (size_t)(HIDDEN * LATENT + LATENT * HIDDEN + HIDDEN + LATENT) * sizeof(float);
    const size_t jac_smem = (size_t)(LATENT * W2P + HIDDEN * W1P + NWAVE * HIDDEN + NWAVE * LATENT * JP) * sizeof(float);

    const float* tin = z;
    float* bufs[2] = { thA, thB };
    for (int i = 0; i < LAYERS; ++i) {
        float* tout = bufs[i & 1];
        fwd_kernel<<<BATCH / 256, 256, fwd_smem, stream>>>(
            tin, W1 + (size_t)i * HIDDEN * LATENT, b1 + (size_t)i * HIDDEN,
            W2 + (size_t)i * LATENT * HIDDEN, b2 + (size_t)i * LATENT,
            tout, msk);
        jac_kernel<<<BATCH / (NWAVE * BPW), NWAVE * 32, jac_smem, stream>>>(
            msk, W1 + (size_t)i * HIDDEN * LATENT, W2 + (size_t)i * LATENT * HIDDEN, lab);
        lse_kernel<<<1, 256, 0, stream>>>(lab, ldj, i);
        tin = tout;
    }
    out_kernel<<<(BATCH * LATENT + 255) / 256, 256, 0, stream>>>(
        bufs[(LAYERS - 1) & 1], ldj, (float*)d_out);
}